// DeltaAttention_40845138985318
// MI455X (gfx1250) — compile-verified
//
#include <hip/hip_runtime.h>
#include <hip/hip_bf16.h>

typedef __attribute__((ext_vector_type(2))) float v2f;
typedef __attribute__((ext_vector_type(8))) float v8f;

#define DDIM   128   // embedding dim
#define LCH    64    // rows (timesteps) per chunk
#define NCHUNK 64    // B * C
#define CPB    32    // chunks per batch

// ---------------- WMMA helpers (fp32 16x16x4) ----------------

static __device__ __forceinline__ v8f wmma4(v2f a, v2f b, v8f c) {
  // 8 args: (neg_a, A, neg_b, B, c_mod, C, reuse_a, reuse_b)
  return __builtin_amdgcn_wmma_f32_16x16x4_f32(false, a, false, b, (short)0, c,
                                               false, false);
}

// A fragment from row-major A: A[row0+m][k..k+3]; m=lane&15, k-pair from lane[4]
static __device__ __forceinline__ v2f ldA(const float* p, int ld, int row0,
                                          int k, int lane) {
  int m = lane & 15, ko = (lane >> 4) << 1;
  const float* q = p + (size_t)(row0 + m) * ld + k + ko;
  v2f a; a.x = q[0]; a.y = q[1];
  return a;
}
// A fragment from column-major source: A[m][kk] = p[(k+kk)*ld + row0+m] (* sgn)
static __device__ __forceinline__ v2f ldAT(const float* p, int ld, int row0,
                                           int k, int lane, float sgn) {
  int m = lane & 15, ko = (lane >> 4) << 1;
  v2f a;
  a.x = sgn * p[(size_t)(k + ko) * ld + row0 + m];
  a.y = sgn * p[(size_t)(k + ko + 1) * ld + row0 + m];
  return a;
}
// B fragment from row-major B (K x N): B[k+kk][col0+n]
static __device__ __forceinline__ v2f ldB(const float* p, int ld, int col0,
                                          int k, int lane) {
  int n = lane & 15, ko = (lane >> 4) << 1;
  v2f b;
  b.x = p[(size_t)(k + ko) * ld + col0 + n];
  b.y = p[(size_t)(k + ko + 1) * ld + col0 + n];
  return b;
}
// B fragment = M^T with M row-major (N x K): B[kk][n] = p[(col0+n)*ld + k+kk]
static __device__ __forceinline__ v2f ldBT(const float* p, int ld, int col0,
                                           int k, int lane) {
  int n = lane & 15, ko = (lane >> 4) << 1;
  const float* q = p + (size_t)(col0 + n) * ld + k + ko;
  v2f b; b.x = q[0]; b.y = q[1];
  return b;
}
// C/D fragment: VGPR r -> row (r + 8*lane[4]), col lane&15
static __device__ __forceinline__ void stC(float* p, int ld, int row0, int col0,
                                           int lane, v8f c) {
  int n = lane & 15, hi = (lane >> 4) * 8;
#pragma unroll
  for (int r = 0; r < 8; ++r)
    p[(size_t)(row0 + r + hi) * ld + col0 + n] = c[r];
}
static __device__ __forceinline__ v8f ldCfrag(const float* p, int ld, int row0,
                                              int col0, int lane) {
  int n = lane & 15, hi = (lane >> 4) * 8;
  v8f c;
#pragma unroll
  for (int r = 0; r < 8; ++r)
    c[r] = p[(size_t)(row0 + r + hi) * ld + col0 + n];
  return c;
}
// masked store for triangular tiles: mode 1 = strict lower (n<m), 2 = incl (n<=m)
static __device__ __forceinline__ void stC_mask(float* p, int ld, int row0,
                                                int col0, int lane, v8f c,
                                                int mode) {
  int n = lane & 15, hi = (lane >> 4) * 8;
#pragma unroll
  for (int r = 0; r < 8; ++r) {
    int m = r + hi;
    float vv = c[r];
    if (mode == 1 && !(n < m)) vv = 0.f;
    if (mode == 2 && !(n <= m)) vv = 0.f;
    p[(size_t)(row0 + m) * ld + col0 + n] = vv;
  }
}

// ---- CDNA5 async memory->LDS staging (GLOBAL_LOAD_ASYNC_TO_LDS_B128) ----
// Copies a 64x128 f32 tile (32 KB) into LDS; tracked by ASYNCcnt, no VGPR data.
static __device__ __forceinline__ void async_copy_chunk(float* lds_dst,
                                                        const float* gsrc,
                                                        int tid) {
  unsigned lds_base = (unsigned)(uintptr_t)lds_dst;  // addrspace(3) byte offset
#pragma unroll
  for (int i = 0; i < 16; ++i) {
    int idx = tid + i * 128;                 // 16-byte packet index, 0..2047
    unsigned lds_addr = lds_base + idx * 16;
    unsigned voff = idx * 16;                // byte offset from SGPR base
    asm volatile("global_load_async_to_lds_b128 %0, %1, %2 offset:0"
                 :
                 : "v"(lds_addr), "v"(voff), "s"(gsrc)
                 : "memory");
  }
}
static __device__ __forceinline__ void async_wait0() {
  asm volatile("s_wait_asynccnt 0x0" ::: "memory");
}

// ---------------- Kernel 1: fused q/k/v/beta_raw projections ----------------
// grid (256 row-tiles, 32): y>>3 selects matrix, y&7 selects 16-col tile.
__global__ void proj_kernel(const float* __restrict__ x,
                            const float* __restrict__ Wq,
                            const float* __restrict__ Wk,
                            const float* __restrict__ Wv,
                            const float* __restrict__ Wb,
                            float* __restrict__ q, float* __restrict__ k,
                            float* __restrict__ v, float* __restrict__ braw) {
  int lane = threadIdx.x & 31;
  int row0 = blockIdx.x * 16;
  int mat = blockIdx.y >> 3;
  int col0 = (blockIdx.y & 7) * 16;
  const float* W = (mat == 0) ? Wq : (mat == 1) ? Wk : (mat == 2) ? Wv : Wb;
  float* out = (mat == 0) ? q : (mat == 1) ? k : (mat == 2) ? v : braw;
  v8f acc = {};
#pragma unroll
  for (int kk = 0; kk < DDIM; kk += 4) {
    v2f a = ldA(x, DDIM, row0, kk, lane);
    v2f b = ldB(W, DDIM, col0, kk, lane);
    acc = wmma4(a, b, acc);
  }
  stC(out, DDIM, row0, col0, lane, acc);
}

// ---------------- Kernel 2: bk = sigmoid(braw) * k ----------------
__global__ void bk_kernel(const float* __restrict__ k,
                          const float* __restrict__ braw,
                          float* __restrict__ bk, int n) {
  int i = blockIdx.x * blockDim.x + threadIdx.x;
  if (i < n) {
    float b = 1.f / (1.f + __expf(-braw[i]));
    bk[i] = b * k[i];
  }
}

// ---------------- Kernel 3: per-chunk WY transform ----------------
// One workgroup (4 waves) per chunk. Produces S_final, W=(I+M)^-1 K, out_intra.
// K and BK tiles (highest reuse: ~3.5x and ~13x) are staged into LDS via
// CDNA5 async memory->LDS loads. Total static LDS = 128 KB (< 320 KB/WGP).
__global__ void chunk_kernel(const float* __restrict__ gq,
                             const float* __restrict__ gk,
                             const float* __restrict__ gv,
                             const float* __restrict__ gbk,
                             float* __restrict__ gSf, float* __restrict__ gW,
                             float* __restrict__ gAttn) {
  __shared__ float sK[64 * 128];    // 32 KB: K chunk (async staged)
  __shared__ float sBK[64 * 128];   // 32 KB: beta*K chunk (async staged)
  __shared__ float sM[64 * 64];     // 16 KB: M = strict_tril(K BK^T)
  __shared__ float sG[64 * 64];     // 16 KB: G = tril(Q BK^T)
  __shared__ float sU[64 * 128];    // 32 KB: U, then reused for W
  int chunk = blockIdx.x;
  size_t base = (size_t)chunk * LCH * DDIM;
  int tid = threadIdx.x, wave = tid >> 5, lane = tid & 31;

  // 0. Async-stage K and BK into LDS (no VGPR round trip, ASYNCcnt tracked).
  async_copy_chunk(sK, gk + base, tid);
  async_copy_chunk(sBK, gbk + base, tid);
  async_wait0();
  __syncthreads();

  // 1. M = strict-lower(K BK^T), zeros elsewhere
  for (int t = wave; t < 16; t += 4) {
    int i = t >> 2, j = t & 3;
    if (j > i) {
      int n = lane & 15, hi = (lane >> 4) * 8;
#pragma unroll
      for (int r = 0; r < 8; ++r) sM[(i * 16 + r + hi) * 64 + j * 16 + n] = 0.f;
    } else {
      v8f acc = {};
#pragma unroll
      for (int kk = 0; kk < DDIM; kk += 4) {
        v2f a = ldA(sK, DDIM, i * 16, kk, lane);
        v2f b = ldBT(sBK, DDIM, j * 16, kk, lane);
        acc = wmma4(a, b, acc);
      }
      stC_mask(sM, 64, i * 16, j * 16, lane, acc, (i == j) ? 1 : 0);
    }
  }
  __syncthreads();

  // 2. Solve (I+M) U = V. Column-parallel forward substitution (col = tid).
  {
    int col = tid;
    for (int t = 0; t < LCH; ++t) {
      float acc = gv[base + (size_t)t * DDIM + col];
      for (int s = 0; s < t; ++s) acc -= sM[t * 64 + s] * sU[s * DDIM + col];
      sU[t * DDIM + col] = acc;
    }
  }
  __syncthreads();

  // 3. G = inclusive-lower(Q BK^T)
  for (int t = wave; t < 16; t += 4) {
    int i = t >> 2, j = t & 3;
    if (j > i) {
      int n = lane & 15, hi = (lane >> 4) * 8;
#pragma unroll
      for (int r = 0; r < 8; ++r) sG[(i * 16 + r + hi) * 64 + j * 16 + n] = 0.f;
    } else {
      v8f acc = {};
#pragma unroll
      for (int kk = 0; kk < DDIM; kk += 4) {
        v2f a = ldA(gq + base, DDIM, i * 16, kk, lane);
        v2f b = ldBT(sBK, DDIM, j * 16, kk, lane);
        acc = wmma4(a, b, acc);
      }
      stC_mask(sG, 64, i * 16, j * 16, lane, acc, (i == j) ? 2 : 0);
    }
  }
  __syncthreads();

  // 4. S_final = BK^T @ U  (128x128, K=64)
  for (int t = wave; t < 64; t += 4) {
    int i = t >> 3, j = t & 7;
    v8f acc = {};
#pragma unroll
    for (int kk = 0; kk < LCH; kk += 4) {
      v2f a = ldAT(sBK, DDIM, i * 16, kk, lane, 1.f);
      v2f b = ldB(sU, DDIM, j * 16, kk, lane);
      acc = wmma4(a, b, acc);
    }
    stC(gSf + (size_t)chunk * DDIM * DDIM, DDIM, i * 16, j * 16, lane, acc);
  }
  // 5. out_intra = G @ U  (64x128, K=64)
  for (int t = wave; t < 32; t += 4) {
    int i = t >> 3, j = t & 7;
    v8f acc = {};
#pragma unroll
    for (int kk = 0; kk < LCH; kk += 4) {
      v2f a = ldA(sG, 64, i * 16, kk, lane);
      v2f b = ldB(sU, DDIM, j * 16, kk, lane);
      acc = wmma4(a, b, acc);
    }
    stC(gAttn + base, DDIM, i * 16, j * 16, lane, acc);
  }
  __syncthreads();

  // 6. Solve (I+M) W = K into sU and write W to global.
  {
    int col = tid;
    for (int t = 0; t < LCH; ++t) {
      float acc = sK[t * DDIM + col];
      for (int s = 0; s < t; ++s) acc -= sM[t * 64 + s] * sU[s * DDIM + col];
      sU[t * DDIM + col] = acc;
      gW[(size_t)chunk * LCH * DDIM + (size_t)t * DDIM + col] = acc;
    }
  }
}

// ---------------- Kernel 4: cross-chunk prefix scan ----------------
// Column-block independent: grid (8 col-blocks, 2 batches), one wave each.
// Replicates reference indexing: emit S, then S = A_prod[c+1] S + S_final[c],
// where A_prod X = X - BK^T (W X).
__global__ void prefix_kernel(const float* __restrict__ gSf,
                              const float* __restrict__ gW,
                              const float* __restrict__ gbk,
                              float* __restrict__ gSp) {
  __shared__ float sS[128 * 16];  // this column block of S (row-major, ld=16)
  __shared__ float sY[64 * 16];   // Y = W_{c+1} @ S_colblock
  int lane = threadIdx.x;
  int nb = blockIdx.x, b = blockIdx.y;
  int col0 = nb * 16;
  for (int i = lane; i < 128 * 16; i += 32) sS[i] = 0.f;
  __syncthreads();
  for (int c = 0; c < CPB; ++c) {
    int chunk = b * CPB + c;
    for (int i = lane; i < 128 * 16; i += 32) {
      int m = i >> 4, n = i & 15;
      gSp[(size_t)chunk * DDIM * DDIM + (size_t)m * DDIM + col0 + n] = sS[i];
    }
    if (c == CPB - 1) break;
    int cn = chunk + 1;
    __syncthreads();
    // Y = W_cn @ S_col (64x16), K=128
#pragma unroll
    for (int yi = 0; yi < 4; ++yi) {
      v8f acc = {};
#pragma unroll
      for (int kk = 0; kk < DDIM; kk += 4) {
        v2f a = ldA(gW + (size_t)cn * LCH * DDIM, DDIM, yi * 16, kk, lane);
        v2f bb = ldB(sS, 16, 0, kk, lane);
        acc = wmma4(a, bb, acc);
      }
      stC(sY, 16, yi * 16, 0, lane, acc);
    }
    __syncthreads();
    // S = S - BK_cn^T @ Y + Sf_c  (negation folded into A operand load)
#pragma unroll
    for (int mi = 0; mi < 8; ++mi) {
      int n = lane & 15, hi = (lane >> 4) * 8;
      v8f acc;
#pragma unroll
      for (int r = 0; r < 8; ++r) {
        int m = mi * 16 + r + hi;
        acc[r] = sS[m * 16 + n] +
                 gSf[(size_t)chunk * DDIM * DDIM + (size_t)m * DDIM + col0 + n];
      }
#pragma unroll
      for (int kk = 0; kk < LCH; kk += 4) {
        v2f a = ldAT(gbk + (size_t)cn * LCH * DDIM, DDIM, mi * 16, kk, lane, -1.f);
        v2f bb = ldB(sY, 16, 0, kk, lane);
        acc = wmma4(a, bb, acc);
      }
#pragma unroll
      for (int r = 0; r < 8; ++r) {
        int m = mi * 16 + r + hi;
        sS[m * 16 + n] = acc[r];
      }
    }
    __syncthreads();
  }
}

// ---------------- Kernel 5: out += Q @ S_prefix ----------------
__global__ void inter_kernel(const float* __restrict__ gq,
                             const float* __restrict__ gSp,
                             float* __restrict__ gAttn) {
  int lane = threadIdx.x;
  int chunk = blockIdx.x;
  int mi = blockIdx.y >> 3, ni = blockIdx.y & 7;
  size_t base = (size_t)chunk * LCH * DDIM;
  v8f acc = ldCfrag(gAttn + base, DDIM, mi * 16, ni * 16, lane);
#pragma unroll
  for (int kk = 0; kk < DDIM; kk += 4) {
    v2f a = ldA(gq + base, DDIM, mi * 16, kk, lane);
    v2f b = ldB(gSp + (size_t)chunk * DDIM * DDIM, DDIM, ni * 16, kk, lane);
    acc = wmma4(a, b, acc);
  }
  stC(gAttn + base, DDIM, mi * 16, ni * 16, lane, acc);
}

// ---------------- Kernel 6: final output projection @ Wo ----------------
__global__ void outproj_kernel(const float* __restrict__ gAttn,
                               const float* __restrict__ Wo,
                               float* __restrict__ out) {
  int lane = threadIdx.x;
  int row0 = blockIdx.x * 16, col0 = blockIdx.y * 16;
  v8f acc = {};
#pragma unroll
  for (int kk = 0; kk < DDIM; kk += 4) {
    v2f a = ldA(gAttn, DDIM, row0, kk, lane);
    v2f b = ldB(Wo, DDIM, col0, kk, lane);
    acc = wmma4(a, b, acc);
  }
  stC(out, DDIM, row0, col0, lane, acc);
}

extern "C" void kernel_launch(void* const* d_in, const int* in_sizes, int n_in,
                              void* d_out, int out_size, void* d_ws,
                              size_t ws_size, hipStream_t stream) {
  const float* x = (const float*)d_in[0];
  const float* Wq = (const float*)d_in[1];
  const float* Wk = (const float*)d_in[2];
  const float* Wv = (const float*)d_in[3];
  const float* Wo = (const float*)d_in[4];
  const float* Wb = (const float*)d_in[5];
  float* out = (float*)d_out;

  const size_t NTOK = 4096;        // B*T
  const size_t ND = NTOK * DDIM;   // 524288 floats
  float* ws = (float*)d_ws;
  float* q = ws;    ws += ND;
  float* k = ws;    ws += ND;
  float* v = ws;    ws += ND;
  float* braw = ws; ws += ND;
  float* bk = ws;   ws += ND;
  float* Sf = ws;   ws += (size_t)NCHUNK * DDIM * DDIM;
  float* W = ws;    ws += (size_t)NCHUNK * LCH * DDIM;
  float* Sp = ws;   ws += (size_t)NCHUNK * DDIM * DDIM;
  float* attn = ws; ws += ND;

  proj_kernel<<<dim3(NTOK / 16, 32), 32, 0, stream>>>(x, Wq, Wk, Wv, Wb, q, k,
                                                      v, braw);
  bk_kernel<<<dim3((unsigned)((ND + 255) / 256)), 256, 0, stream>>>(
      k, braw, bk, (int)ND);
  chunk_kernel<<<dim3(NCHUNK), 128, 0, stream>>>(q, k, v, bk, Sf, W, attn);
  prefix_kernel<<<dim3(8, 2), 32, 0, stream>>>(Sf, W, bk, Sp);
  inter_kernel<<<dim3(NCHUNK, 32), 32, 0, stream>>>(q, Sp, attn);
  outproj_kernel<<<dim3(NTOK / 16, 8), 32, 0, stream>>>(attn, Wo, out);
}